// MyNet_3496103379504
// MI455X (gfx1250) — compile-verified
//
#include <hip/hip_runtime.h>

#define NN 100000      // nodes
#define NE 3200000     // edges (before self loops)
#define FIN 128
#define HD 16
#define NG 1000

typedef float v2f __attribute__((ext_vector_type(2)));
typedef float v8f __attribute__((ext_vector_type(8)));

// ---------- utility fills ----------
__global__ void fill_kernel(float* __restrict__ p, int n, float v) {
    int i = blockIdx.x * blockDim.x + threadIdx.x;
    if (i < n) p[i] = v;
}

// ---------- degree: deg starts at 1.0 (self loop), add 1 per incoming edge ----------
__global__ void degree_kernel(const int* __restrict__ dst, float* __restrict__ deg) {
    int e = blockIdx.x * blockDim.x + threadIdx.x;
    if (e < NE) atomicAdd(&deg[dst[e]], 1.0f);
}

__global__ void rsqrt_kernel(float* __restrict__ deg) {
    int i = blockIdx.x * blockDim.x + threadIdx.x;
    if (i < NN) deg[i] = rsqrtf(deg[i]);   // deg >= 1 always (self loop)
}

// ---------- h_pre = x @ W1  via V_WMMA_F32_16X16X4_F32 ----------
// One wave computes a 16x16 output tile (16 rows of x, all 16 cols of W1).
// N = 100000 = 6250 * 16 exactly, so no row remainder.
__global__ void gemm1_wmma(const float* __restrict__ x, const float* __restrict__ W1,
                           float* __restrict__ h_pre) {
    int wave = (blockIdx.x * blockDim.x + threadIdx.x) >> 5;
    int lane = threadIdx.x & 31;
    if (wave >= NN / 16) return;           // whole-wave uniform exit -> EXEC all-1 inside

    const int m    = lane & 15;            // row within tile (also col index for B)
    const int koff = (lane >> 4) << 1;     // lanes 0-15 -> K+{0,1}; lanes 16-31 -> K+{2,3}
    const float* xr = x + (size_t)(wave * 16 + m) * FIN;

    v8f c = {};
#pragma unroll
    for (int k0 = 0; k0 < FIN; k0 += 4) {
        v2f a, b;
        // A 16x4 f32 layout (ISA 7.12.2): VGPR0 = K=k0 / k0+2, VGPR1 = K=k0+1 / k0+3
        a.x = xr[k0 + koff];
        a.y = xr[k0 + koff + 1];
        // B 4x16 f32, row-major W1[k][n], n striped on lanes, K split like A
        b.x = W1[(k0 + koff) * HD + m];
        b.y = W1[(k0 + koff + 1) * HD + m];
        c = __builtin_amdgcn_wmma_f32_16x16x4_f32(
            /*neg_a=*/false, a, /*neg_b=*/false, b,
            /*c_mod=*/(short)0, c, /*reuse_a=*/false, /*reuse_b=*/false);
    }

    // D layout: VGPR r -> M=r (lanes 0-15), M=r+8 (lanes 16-31); N = lane&15
    const int n     = lane & 15;
    const int mbase = (lane >> 4) * 8;
#pragma unroll
    for (int r = 0; r < 8; ++r)
        h_pre[(size_t)(wave * 16 + mbase + r) * HD + n] = c[r];
}

// ---------- layer-1 scatter: agg1[dst,:] += h_pre[src,:] * dinv[src]*dinv[dst] ----------
// One thread per (edge, feature); self loops are edges [NE, NE+NN).
__global__ void edge_agg16(const int* __restrict__ src, const int* __restrict__ dst,
                           const float* __restrict__ dinv,
                           const float* __restrict__ h, float* __restrict__ agg) {
    unsigned gid = blockIdx.x * blockDim.x + threadIdx.x;
    unsigned e = gid >> 4;
    int f = (int)(gid & 15u);
    if (e >= (unsigned)(NE + NN)) return;
    int s, d;
    if (e < (unsigned)NE) { s = src[e]; d = dst[e]; }
    else                  { s = d = (int)(e - NE); }
    float nm = dinv[s] * dinv[d];
    atomicAdd(&agg[(size_t)d * HD + f], h[(size_t)s * HD + f] * nm);
}

// ---------- h1 = relu(agg1 + b1) in place ----------
__global__ void bias_relu(float* __restrict__ agg, const float* __restrict__ b1) {
    int i = blockIdx.x * blockDim.x + threadIdx.x;
    if (i < NN * HD) {
        float v = agg[i] + b1[i & 15];
        agg[i] = v > 0.0f ? v : 0.0f;
    }
}

// ---------- h2_pre[n] = h1[n,:] . W2 ----------
__global__ void dot16_kernel(const float* __restrict__ h1, const float* __restrict__ W2,
                             float* __restrict__ h2) {
    int nidx = blockIdx.x * blockDim.x + threadIdx.x;
    if (nidx >= NN) return;
    const float4* hv = (const float4*)(h1 + (size_t)nidx * HD);
    const float4* wv = (const float4*)W2;
    float s = 0.0f;
#pragma unroll
    for (int q = 0; q < 4; ++q) {
        float4 a = hv[q], b = wv[q];
        s += a.x * b.x + a.y * b.y + a.z * b.z + a.w * b.w;
    }
    h2[nidx] = s;
}

// ---------- layer-2 scatter: agg2[dst] += h2[src] * norm ----------
__global__ void edge_agg1(const int* __restrict__ src, const int* __restrict__ dst,
                          const float* __restrict__ dinv,
                          const float* __restrict__ h2, float* __restrict__ agg2) {
    int e = blockIdx.x * blockDim.x + threadIdx.x;
    if (e >= NE + NN) return;
    int s, d;
    if (e < NE) { s = src[e]; d = dst[e]; }
    else        { s = d = e - NE; }
    atomicAdd(&agg2[d], h2[s] * dinv[s] * dinv[d]);
}

// ---------- global add pool: out[batch[n]] += agg2[n] + b2 ----------
__global__ void pool_kernel(const float* __restrict__ agg2, const int* __restrict__ batch,
                            const float* __restrict__ b2, float* __restrict__ out) {
    int n = blockIdx.x * blockDim.x + threadIdx.x;
    if (n < NN) atomicAdd(&out[batch[n]], agg2[n] + b2[0]);
}

extern "C" void kernel_launch(void* const* d_in, const int* in_sizes, int n_in,
                              void* d_out, int out_size, void* d_ws, size_t ws_size,
                              hipStream_t stream) {
    const float* x    = (const float*)d_in[0];   // [NN, 128]
    const int*   ei   = (const int*)d_in[1];     // [2, NE]
    const int*   bat  = (const int*)d_in[2];     // [NN]
    const float* W1   = (const float*)d_in[3];   // [128, 16]
    const float* b1   = (const float*)d_in[4];   // [16]
    const float* W2   = (const float*)d_in[5];   // [16, 1]
    const float* b2   = (const float*)d_in[6];   // [1]
    float* out = (float*)d_out;                  // [NG]

    const int* src = ei;
    const int* dst = ei + NE;

    // workspace layout (floats): dinv | h_pre | agg1 | h2 | agg2  = 35*NN floats (~14 MB)
    float* ws   = (float*)d_ws;
    float* dinv = ws;
    float* hpre = dinv + NN;
    float* agg1 = hpre + (size_t)NN * HD;
    float* h2   = agg1 + (size_t)NN * HD;
    float* agg2 = h2 + NN;

    const int B = 256;

    // init
    fill_kernel<<<(NN + B - 1) / B, B, 0, stream>>>(dinv, NN, 1.0f);          // deg = 1 (self loop)
    fill_kernel<<<(NN * HD + B - 1) / B, B, 0, stream>>>(agg1, NN * HD, 0.0f);
    fill_kernel<<<(NN + B - 1) / B, B, 0, stream>>>(agg2, NN, 0.0f);
    fill_kernel<<<(NG + B - 1) / B, B, 0, stream>>>(out, NG, 0.0f);

    // degree + D^{-1/2}
    degree_kernel<<<(NE + B - 1) / B, B, 0, stream>>>(dst, dinv);
    rsqrt_kernel<<<(NN + B - 1) / B, B, 0, stream>>>(dinv);

    // layer 1: dense (WMMA), then edge scatter, then bias+relu
    {
        int waves = NN / 16;                       // 6250
        int wavesPerBlock = 4;                     // 128 threads
        int blocks = (waves + wavesPerBlock - 1) / wavesPerBlock;
        gemm1_wmma<<<blocks, wavesPerBlock * 32, 0, stream>>>(x, W1, hpre);
    }
    {
        unsigned total = (unsigned)(NE + NN) * 16u;  // 52.8M threads
        edge_agg16<<<(total + B - 1) / B, B, 0, stream>>>(src, dst, dinv, hpre, agg1);
    }
    bias_relu<<<(NN * HD + B - 1) / B, B, 0, stream>>>(agg1, b1);

    // layer 2: per-node dot, edge scatter
    dot16_kernel<<<(NN + B - 1) / B, B, 0, stream>>>(agg1, W2, h2);
    edge_agg1<<<(NE + NN + B - 1) / B, B, 0, stream>>>(src, dst, dinv, h2, agg2);

    // pool (+b2 per node, matching reference which adds bias before pooling)
    pool_kernel<<<(NN + B - 1) / B, B, 0, stream>>>(agg2, bat, b2, out);
}